// FeaturePropagation_19481971655382
// MI455X (gfx1250) — compile-verified
//
#include <hip/hip_runtime.h>

typedef __attribute__((ext_vector_type(2))) float v2f;
typedef __attribute__((ext_vector_type(8))) float v8f;

#define NVERT   100000
#define NCENT   2048
#define NDIM    64
#define CHID    128
#define COUT    128
#define NSTRIP  (NVERT / 16)   // 6250 exactly
#define BN_EPS  1e-5f

// K-pair swizzle for WMMA B operands: element (k, col) of a [K,128] matrix is
// stored at (k>>1)*256 + col*2 + (k&1), so the two K-values a lane feeds into
// one v_wmma B register pair are adjacent 8-byte-aligned floats (one b64 load).
__device__ __forceinline__ int bswz(int k, int col) {
    return ((k >> 1) << 8) + (col << 1) + (k & 1);
}

// ---------------------------------------------------------------------------
// Kernel 0: zero the global stat accumulators (256 floats)
// ---------------------------------------------------------------------------
__global__ void zero_stats_kernel(float* __restrict__ stats) {
    stats[threadIdx.x] = 0.0f;
}

// ---------------------------------------------------------------------------
// Kernel 0b: pre-swizzle W2 [128,128] into K-pair layout in workspace.
// ---------------------------------------------------------------------------
__global__ void swizzle_w2_kernel(const float* __restrict__ W2,
                                  float* __restrict__ W2s) {
    for (int i = threadIdx.x + blockIdx.x * 256; i < CHID * COUT;
         i += 256 * gridDim.x) {
        const int k = i >> 7, col = i & 127;
        W2s[bswz(k, col)] = W2[i];
    }
}

// ---------------------------------------------------------------------------
// Kernel 1: IDW interpolation.  One thread per vertex; centroids in LDS
// (padded float4 -> uniform-address broadcast ds_load_b128, no conflicts).
// P==2 so the weight is exactly 1/d^2: top-3 selection runs on d^2, no sqrt.
// ---------------------------------------------------------------------------
__global__ __launch_bounds__(256) void idw_kernel(
    const float* __restrict__ vertices,          // [N,3]
    const float* __restrict__ centroids,         // [M,3]
    const float* __restrict__ feats,             // [M,64]
    float* __restrict__ interp)                  // [N,64]
{
    __shared__ float4 c_lds[NCENT];              // 32 KB
    for (int m = threadIdx.x; m < NCENT; m += 256) {
        float4 c;
        c.x = centroids[3 * m + 0];
        c.y = centroids[3 * m + 1];
        c.z = centroids[3 * m + 2];
        c.w = 0.0f;
        c_lds[m] = c;
    }
    __syncthreads();

    const int v = blockIdx.x * 256 + threadIdx.x;
    const bool active = (v < NVERT);
    float vx = 0.f, vy = 0.f, vz = 0.f;
    if (active) {
        vx = vertices[3 * v + 0];
        vy = vertices[3 * v + 1];
        vz = vertices[3 * v + 2];
    }

    float t0 = 3.0e38f, t1 = 3.0e38f, t2 = 3.0e38f;   // top-3 smallest d^2
    int   i0 = 0,       i1 = 0,       i2 = 0;

    #pragma unroll 4
    for (int m = 0; m < NCENT; ++m) {
        float4 c = c_lds[m];
        float dx = vx - c.x;
        float dy = vy - c.y;
        float dz = vz - c.z;
        float d2 = fmaf(dx, dx, fmaf(dy, dy, dz * dz));
        if (d2 < t2) {
            if (d2 < t1) {
                t2 = t1; i2 = i1;
                if (d2 < t0) { t1 = t0; i1 = i0; t0 = d2; i0 = m; }
                else         { t1 = d2; i1 = m; }
            } else { t2 = d2; i2 = m; }
        }
    }

    const float4* f0 = (const float4*)(feats + (size_t)i0 * NDIM);
    const float4* f1 = (const float4*)(feats + (size_t)i1 * NDIM);
    const float4* f2 = (const float4*)(feats + (size_t)i2 * NDIM);
    // hide L2 gather latency behind the weight math (global_prefetch_b8)
    __builtin_prefetch(f0, 0, 0);
    __builtin_prefetch(f1, 0, 0);
    __builtin_prefetch(f2, 0, 0);

    // weights: all nonzero -> 1/d^2 ; any exact hit -> 1 for hits, 0 otherwise
    const bool z0 = (t0 == 0.f), z1 = (t1 == 0.f), z2 = (t2 == 0.f);
    const bool anyz = z0 || z1 || z2;
    float w0 = anyz ? (z0 ? 1.f : 0.f) : (1.0f / t0);
    float w1 = anyz ? (z1 ? 1.f : 0.f) : (1.0f / t1);
    float w2 = anyz ? (z2 ? 1.f : 0.f) : (1.0f / t2);
    const float inv = 1.0f / (w0 + w1 + w2);
    w0 *= inv; w1 *= inv; w2 *= inv;

    if (active) {
        float4* dst = (float4*)(interp + (size_t)v * NDIM);
        #pragma unroll
        for (int d = 0; d < NDIM / 4; ++d) {
            float4 a = f0[d], b = f1[d], c = f2[d];
            float4 r;
            r.x = fmaf(w0, a.x, fmaf(w1, b.x, w2 * c.x));
            r.y = fmaf(w0, a.y, fmaf(w1, b.y, w2 * c.y));
            r.z = fmaf(w0, a.z, fmaf(w1, b.z, w2 * c.z));
            r.w = fmaf(w0, a.w, fmaf(w1, b.w, w2 * c.w));
            dst[d] = r;
        }
    }
}

// ---------------------------------------------------------------------------
// Kernel 2: h = interp @ W1[0:64,:] + b1  via V_WMMA_F32_16X16X4_F32,
// plus per-channel sum / sum-of-squares (LDS ds_add_f32 reduce -> global).
// One wave owns a 16-row x 128-col strip (8 v8f accumulators).  W1 sits in
// LDS in K-pair-swizzled form so each B operand is a single ds_load_b64.
// ---------------------------------------------------------------------------
__global__ __launch_bounds__(256) void gemm1_stats_kernel(
    const float* __restrict__ interp,            // [N,64]
    const float* __restrict__ W1,                // [67,128] row-major (rows>=64 dead)
    const float* __restrict__ b1,                // [128]
    float* __restrict__ h,                       // [N,128]
    float* __restrict__ gsum,                    // [128]
    float* __restrict__ gsumsq)                  // [128]
{
    __shared__ __align__(16) float w_lds[NDIM * CHID];   // 32 KB, swizzled
    __shared__ float b_lds[CHID];
    __shared__ float s_sum[CHID];
    __shared__ float s_sq[CHID];

    for (int i = threadIdx.x; i < NDIM * CHID; i += 256) {
        const int k = i >> 7, col = i & 127;
        w_lds[bswz(k, col)] = W1[i];
    }
    if (threadIdx.x < CHID) {
        b_lds[threadIdx.x] = b1[threadIdx.x];
        s_sum[threadIdx.x] = 0.0f;
        s_sq[threadIdx.x]  = 0.0f;
    }
    __syncthreads();

    const int wave  = threadIdx.x >> 5;
    const int lane  = threadIdx.x & 31;
    const int strip = blockIdx.x * 8 + wave;

    if (strip < NSTRIP) {                        // wave-uniform: EXEC all-1s inside
        const int mrow = lane & 15;              // A/C column-lane mapping
        const int kb   = (lane >> 4) * 2;        // K pair held by this half-wave
        const size_t rowA = (size_t)(strip * 16 + mrow) * NDIM;

        v8f acc[8];
        #pragma unroll
        for (int t = 0; t < 8; ++t) acc[t] = (v8f)0.0f;

        for (int k0 = 0; k0 < NDIM; k0 += 4) {
            const int ka = k0 + kb;              // even
            const v2f a = *(const v2f*)(interp + rowA + ka);
            const int bbase = (ka >> 1) << 8;    // swizzled K-pair row
            #pragma unroll
            for (int t = 0; t < 8; ++t) {
                const int col = t * 16 + mrow;
                const v2f b = *(const v2f*)&w_lds[bbase + (col << 1)];
                acc[t] = __builtin_amdgcn_wmma_f32_16x16x4_f32(
                    false, a, false, b, (short)0, acc[t], false, false);
            }
        }

        const int rowoff = (lane >> 4) * 8;      // C/D: row = g + 8*(lane/16)
        #pragma unroll
        for (int t = 0; t < 8; ++t) {
            const int col  = t * 16 + mrow;
            const float bi = b_lds[col];
            float s = 0.f, sq = 0.f;
            #pragma unroll
            for (int g = 0; g < 8; ++g) {
                const float val = acc[t][g] + bi;
                h[(size_t)(strip * 16 + rowoff + g) * CHID + col] = val;
                s  += val;
                sq  = fmaf(val, val, sq);
            }
            atomicAdd(&s_sum[col], s);
            atomicAdd(&s_sq[col],  sq);
        }
    }
    __syncthreads();
    if (threadIdx.x < CHID) {
        atomicAdd(&gsum[threadIdx.x],   s_sum[threadIdx.x]);
        atomicAdd(&gsumsq[threadIdx.x], s_sq[threadIdx.x]);
    }
}

// ---------------------------------------------------------------------------
// Kernel 3: fold stats into affine BN:  y = h*scale + shift
// ---------------------------------------------------------------------------
__global__ void bn_finalize_kernel(
    const float* __restrict__ gsum, const float* __restrict__ gsumsq,
    const float* __restrict__ gamma, const float* __restrict__ beta,
    float* __restrict__ scale, float* __restrict__ shift)
{
    const int c = threadIdx.x;
    if (c < CHID) {
        const float invN = 1.0f / (float)NVERT;
        const float mu   = gsum[c] * invN;
        const float var  = fmaf(-mu, mu, gsumsq[c] * invN);
        const float sc   = gamma[c] * rsqrtf(var + BN_EPS);
        scale[c] = sc;
        shift[c] = fmaf(-mu, sc, beta[c]);
    }
}

// ---------------------------------------------------------------------------
// Kernel 4: out = relu(h*scale+shift) @ W2 + b2 (BN+ReLU fused into A load).
// W2 pre-swizzled in ws: each B operand is one coalesced global_load_b64
// (lanes 0..15 cover a contiguous 128B segment); 64 KB stays L1-hot.
// ---------------------------------------------------------------------------
__global__ __launch_bounds__(256) void gemm2_kernel(
    const float* __restrict__ h,                 // [N,128]
    const float* __restrict__ scale,
    const float* __restrict__ shift,
    const float* __restrict__ W2s,               // [128,128] K-pair swizzled
    const float* __restrict__ b2,                // [128]
    float* __restrict__ out)                     // [N,128]
{
    const int wave  = threadIdx.x >> 5;
    const int lane  = threadIdx.x & 31;
    const int strip = blockIdx.x * 8 + wave;
    if (strip >= NSTRIP) return;                 // wave-uniform exit

    const int mrow = lane & 15;
    const int kb   = (lane >> 4) * 2;
    const size_t rowA = (size_t)(strip * 16 + mrow) * CHID;

    v8f acc[8];
    #pragma unroll
    for (int t = 0; t < 8; ++t) acc[t] = (v8f)0.0f;

    for (int k0 = 0; k0 < CHID; k0 += 4) {
        const int ka = k0 + kb;                  // even
        const v2f hp = *(const v2f*)(h + rowA + ka);
        v2f a;
        a.x = fmaxf(fmaf(hp.x, scale[ka],     shift[ka]),     0.0f);
        a.y = fmaxf(fmaf(hp.y, scale[ka + 1], shift[ka + 1]), 0.0f);
        const int bbase = (ka >> 1) << 8;
        #pragma unroll
        for (int t = 0; t < 8; ++t) {
            const int col = t * 16 + mrow;
            const v2f b = *(const v2f*)&W2s[bbase + (col << 1)];
            acc[t] = __builtin_amdgcn_wmma_f32_16x16x4_f32(
                false, a, false, b, (short)0, acc[t], false, false);
        }
    }

    const int rowoff = (lane >> 4) * 8;
    #pragma unroll
    for (int t = 0; t < 8; ++t) {
        const int col  = t * 16 + mrow;
        const float bi = b2[col];
        #pragma unroll
        for (int g = 0; g < 8; ++g)
            out[(size_t)(strip * 16 + rowoff + g) * COUT + col] = acc[t][g] + bi;
    }
}

// ---------------------------------------------------------------------------
extern "C" void kernel_launch(void* const* d_in, const int* in_sizes, int n_in,
                              void* d_out, int out_size, void* d_ws, size_t ws_size,
                              hipStream_t stream) {
    const float* vertices = (const float*)d_in[0];   // [100000,3]
    const float* centroids = (const float*)d_in[1];  // [2048,3]
    const float* feats    = (const float*)d_in[2];   // [2048,64]
    const float* W1       = (const float*)d_in[3];   // [67,128]
    const float* b1       = (const float*)d_in[4];   // [128]
    const float* gamma    = (const float*)d_in[5];   // [128]
    const float* beta     = (const float*)d_in[6];   // [128]
    const float* W2       = (const float*)d_in[7];   // [128,128]
    const float* b2       = (const float*)d_in[8];   // [128]
    float* out = (float*)d_out;                      // [100000,128]

    // workspace layout
    float* interp = (float*)d_ws;                    // 100000*64
    float* hbuf   = interp + (size_t)NVERT * NDIM;   // 100000*128
    float* gsum   = hbuf   + (size_t)NVERT * CHID;   // 128
    float* gsumsq = gsum + CHID;                     // 128
    float* scale  = gsumsq + CHID;                   // 128
    float* shift  = scale + CHID;                    // 128
    float* W2s    = shift + CHID;                    // 128*128 swizzled

    zero_stats_kernel<<<1, 256, 0, stream>>>(gsum);  // zeros gsum+gsumsq (256 floats)
    swizzle_w2_kernel<<<8, 256, 0, stream>>>(W2, W2s);

    idw_kernel<<<(NVERT + 255) / 256, 256, 0, stream>>>(
        vertices, centroids, feats, interp);

    gemm1_stats_kernel<<<(NSTRIP + 7) / 8, 256, 0, stream>>>(
        interp, W1, b1, hbuf, gsum, gsumsq);

    bn_finalize_kernel<<<1, 128, 0, stream>>>(gsum, gsumsq, gamma, beta, scale, shift);

    gemm2_kernel<<<(NSTRIP + 7) / 8, 256, 0, stream>>>(
        hbuf, scale, shift, W2s, b2, out);
}